// YOLOV2Loss_78967268704706
// MI455X (gfx1250) — compile-verified
//
#include <hip/hip_runtime.h>
#include <math.h>

#ifndef __has_builtin
#define __has_builtin(x) 0
#endif

// CDNA5 async global->LDS path (gfx1250). Guarded so host pass and older
// toolchains fall back to plain vector copies and the TU always compiles.
#if __has_builtin(__builtin_amdgcn_global_load_async_to_lds_b128)
#define YV2_HAVE_ASYNC_LDS 1
#else
#define YV2_HAVE_ASYNC_LDS 0
#endif

#define YV2_BOX 5
#define YV2_NC 80
#define YV2_D 85           // 5 + 80
#define YV2_CH 425         // BOX * D
#define YV2_CPT 32         // cells per tile (contiguous 54,400 B per tensor)
#define YV2_BLK 256        // 8 waves (wave32)

#if YV2_HAVE_ASYNC_LDS
// Exact pointee type from the hipcc diagnostic:
//   'int __attribute__((__vector_size__(4 * sizeof(int)))) __device__ *'
typedef int yv2_i4 __attribute__((vector_size(16)));
typedef __attribute__((address_space(1))) yv2_i4* yv2_gptr;  // global (AS1)
typedef __attribute__((address_space(3))) yv2_i4* yv2_lptr;  // LDS (AS3)
#endif

// ---------------------------------------------------------------------------
// Kernel 1: one workgroup per 32-cell tile.
//   Phase A: async-copy tile of preds+targets into LDS (108.8 KB).
//   Phase B: one thread per (cell, box): IoU match, losses, class NLL.
//   Phase C: wave + LDS reduction -> one partial float per workgroup.
// ---------------------------------------------------------------------------
__global__ __launch_bounds__(YV2_BLK) void yolov2_loss_tiles(
    const float* __restrict__ preds, const float* __restrict__ targets,
    float* __restrict__ partial, int ncells) {
  extern __shared__ __align__(16) float sh[];       // [2*CPT*CH + 8]
  float* shp = sh;                                  // CPT*CH floats
  float* sht = sh + YV2_CPT * YV2_CH;               // CPT*CH floats
  float* wred = sh + 2 * YV2_CPT * YV2_CH;          // 8 floats

  const int tid = threadIdx.x;
  const long long c0 = (long long)blockIdx.x * YV2_CPT;
  if (c0 >= ncells) return;  // whole block exits together (no barriers yet)
  const int cells = min(YV2_CPT, (int)(ncells - c0));
  const size_t gbase = (size_t)c0 * YV2_CH;
  const float* gp = preds + gbase;
  const float* gt = targets + gbase;
  const int nfl = cells * YV2_CH;

  // ---- Phase A: stage tile into LDS -------------------------------------
  if (((gbase & 3u) == 0) && ((nfl & 3) == 0)) {
    const int nf4 = nfl >> 2;
    const float4* gp4 = (const float4*)gp;
    const float4* gt4 = (const float4*)gt;
    float4* lp4 = (float4*)shp;
    float4* lt4 = (float4*)sht;
#if YV2_HAVE_ASYNC_LDS
    for (int i = tid; i < nf4; i += YV2_BLK) {
      __builtin_amdgcn_global_load_async_to_lds_b128(
          (yv2_gptr)(gp4 + i), (yv2_lptr)(lp4 + i), 0, 0);
      __builtin_amdgcn_global_load_async_to_lds_b128(
          (yv2_gptr)(gt4 + i), (yv2_lptr)(lt4 + i), 0, 0);
    }
#if __has_builtin(__builtin_amdgcn_s_wait_asynccnt)
    __builtin_amdgcn_s_wait_asynccnt(0);
#else
    asm volatile("s_wait_asynccnt 0" ::: "memory");
#endif
#else
    for (int i = tid; i < nf4; i += YV2_BLK) {
      lp4[i] = gp4[i];
      lt4[i] = gt4[i];
    }
#endif
  } else {  // generic tail path (unused for the 86,528-cell harness shapes)
    for (int i = tid; i < nfl; i += YV2_BLK) {
      shp[i] = gp[i];
      sht[i] = gt[i];
    }
  }
  __syncthreads();

  // ---- Phase B: per-(cell, box) loss ------------------------------------
  float acc = 0.0f;
  const int items = cells * YV2_BOX;
  for (int w = tid; w < items; w += YV2_BLK) {
    const int cell = w / YV2_BOX;
    const int b = w - cell * YV2_BOX;
    const float* P = shp + cell * YV2_CH + b * YV2_D;
    const float* T = sht + cell * YV2_CH;

    const float px = P[0], py = P[1], pw = P[2], ph = P[3], pc = P[4];
    const float px1 = px - pw * 0.5f, px2 = px + pw * 0.5f;
    const float py1 = py - ph * 0.5f, py2 = py + ph * 0.5f;
    const float areap = (px2 - px1) * (py2 - py1);

    // best-IoU target box (first occurrence of max, like jnp.argmax)
    int best = 0;
    float bestv = -INFINITY;
#pragma unroll
    for (int k = 0; k < YV2_BOX; ++k) {
      const float* Tk = T + k * YV2_D;
      const float tx = Tk[0], ty = Tk[1], tw = Tk[2], th = Tk[3];
      const float tx1 = tx - tw * 0.5f, tx2 = tx + tw * 0.5f;
      const float ty1 = ty - th * 0.5f, ty2 = ty + th * 0.5f;
      float iw = fminf(px2, tx2) - fmaxf(px1, tx1);
      iw = fmaxf(iw, 0.0f);
      float ih = fminf(py2, ty2) - fmaxf(py1, ty1);
      ih = fmaxf(ih, 0.0f);
      const float inter = iw * ih;
      const float areat = (tx2 - tx1) * (ty2 - ty1);
      const float iou = inter / (areap + areat - inter + 1e-6f);
      if (iou > bestv) {
        bestv = iou;
        best = k;
      }
    }

    const float* M = T + best * YV2_D;
    const float m0 = M[0], m1 = M[1], m2 = M[2], m3 = M[3], m4 = M[4];
    const float obj = (m4 > 0.0f) ? 1.0f : 0.0f;

    // xy loss
    const float dx = px - m0, dy = py - m1;
    acc += 5.0f * obj * (dx * dx + dy * dy);
    // wh loss
    const float pws = sqrtf(fabsf(pw + 1e-6f)), phs = sqrtf(fabsf(ph + 1e-6f));
    const float mws = sqrtf(fabsf(m2 + 1e-6f)), mhs = sqrtf(fabsf(m3 + 1e-6f));
    const float dw = pws - mws, dh = phs - mhs;
    acc += 5.0f * obj * (dw * dw + dh * dh);
    // confidence loss (obj weight 1, noobj weight 0.5)
    const float dc = pc - m4;
    acc += (obj + 0.5f * (1.0f - obj)) * (dc * dc);

    // class NLL (target class argmax comes from t at the SAME box index b)
    if (obj > 0.0f) {
      const float* PC = P + 5;
      const float* TC = T + b * YV2_D + 5;
      int tcls = 0;
      float tb = TC[0];
      float mx = PC[0];
      for (int k = 1; k < YV2_NC; ++k) {
        const float tv = TC[k];
        if (tv > tb) {
          tb = tv;
          tcls = k;
        }
        mx = fmaxf(mx, PC[k]);
      }
      float se = 0.0f;
      for (int k = 0; k < YV2_NC; ++k) se += expf(PC[k] - mx);
      acc += (mx + logf(se)) - PC[tcls];  // -log_softmax[tcls]
    }
  }

  // ---- Phase C: reduce to one partial per workgroup ----------------------
  for (int o = 16; o > 0; o >>= 1) acc += __shfl_xor(acc, o, 32);
  if ((tid & 31) == 0) wred[tid >> 5] = acc;
  __syncthreads();
  if (tid == 0) {
    float s = 0.0f;
#pragma unroll
    for (int wv = 0; wv < YV2_BLK / 32; ++wv) s += wred[wv];
    partial[blockIdx.x] = s;
  }
}

// ---------------------------------------------------------------------------
// Kernel 2: deterministic single-workgroup sum of the per-tile partials
// (fixed order per launch -> identical result on every graph replay).
// ---------------------------------------------------------------------------
__global__ __launch_bounds__(YV2_BLK) void yolov2_loss_finalize(
    const float* __restrict__ partial, int n, float* __restrict__ out) {
  __shared__ float wred[YV2_BLK / 32];
  float acc = 0.0f;
  for (int i = threadIdx.x; i < n; i += YV2_BLK) acc += partial[i];
  for (int o = 16; o > 0; o >>= 1) acc += __shfl_xor(acc, o, 32);
  if ((threadIdx.x & 31) == 0) wred[threadIdx.x >> 5] = acc;
  __syncthreads();
  if (threadIdx.x == 0) {
    float s = 0.0f;
#pragma unroll
    for (int wv = 0; wv < YV2_BLK / 32; ++wv) s += wred[wv];
    out[0] = s;
  }
}

extern "C" void kernel_launch(void* const* d_in, const int* in_sizes, int n_in,
                              void* d_out, int out_size, void* d_ws,
                              size_t ws_size, hipStream_t stream) {
  (void)n_in;
  (void)out_size;
  (void)ws_size;
  const float* preds = (const float*)d_in[0];
  const float* targets = (const float*)d_in[1];
  float* out = (float*)d_out;
  float* partial = (float*)d_ws;

  const int total = in_sizes[0];          // N*S*S*425
  const int ncells = total / YV2_CH;      // N*S*S = 86,528
  const int ntiles = (ncells + YV2_CPT - 1) / YV2_CPT;  // 2704
  const size_t shbytes = (size_t)(2 * YV2_CPT * YV2_CH + 8) * sizeof(float);

  yolov2_loss_tiles<<<ntiles, YV2_BLK, shbytes, stream>>>(preds, targets,
                                                          partial, ncells);
  yolov2_loss_finalize<<<1, YV2_BLK, 0, stream>>>(partial, ntiles, out);
}